// GIN_90056874262917
// MI455X (gfx1250) — compile-verified
//
#include <hip/hip_runtime.h>
#include <hip/hip_bf16.h>

// ---------------------------------------------------------------------------
// GIN forward on MI455X (gfx1250): fused WMMA-f32 GEMM layers + L2-resident
// atomic aggregation. Matrix math uses V_WMMA_F32_16X16X4_F32 (wave32).
// Round-2 changes: W stored transposed in LDS so A/B operands are single
// aligned ds_load_b64's (no register shuffling), and each wave now computes a
// 32-node x 64-col tile (8 accumulators) to double WMMA work per B fetch.
// ---------------------------------------------------------------------------

#define N_NODES_C  100000
#define N_EDGES_C  1000000
#define N_GRAPHS_C 128
#define HDIM       64

typedef float v2f __attribute__((ext_vector_type(2)));
typedef float v8f __attribute__((ext_vector_type(8)));

static __device__ __forceinline__ v8f wmma4(v2f a, v2f b, v8f c) {
    return __builtin_amdgcn_wmma_f32_16x16x4_f32(false, a, false, b, (short)0, c, false, false);
}

// ---------------------------------------------------------------------------
// Edge scatter-add: agg[dst] += x[src]   (one thread per (edge, 4-col chunk))
// Gathers are float4-coalesced; scatters are f32 atomics that resolve in L2
// (agg buffer << 192MB L2).
// ---------------------------------------------------------------------------
template <int FIN>
__global__ __launch_bounds__(256) void scatter_add_kernel(
    const float* __restrict__ X, const int* __restrict__ src,
    const int* __restrict__ dst, float* __restrict__ Agg)
{
    constexpr int CH = FIN / 4;
    long long tid = (long long)blockIdx.x * blockDim.x + threadIdx.x;
    if (tid >= (long long)N_EDGES_C * CH) return;
    int e = (int)(tid / CH);
    int c = (int)(tid % CH) * 4;
    int s = src[e];
    int d = dst[e];
    const float4 v = *(const float4*)(X + (size_t)s * FIN + c);
    float* p = Agg + (size_t)d * FIN + c;
    atomicAdd(p + 0, v.x);
    atomicAdd(p + 1, v.y);
    atomicAdd(p + 2, v.z);
    atomicAdd(p + 3, v.w);
}

// ---------------------------------------------------------------------------
// Fused GIN layer: Hout = relu( relu( BN( (X+Agg) @ W1 ) ) @ W2 )
// Block = 128 threads = 4 waves; each wave computes a 32-node x 64-col tile
// (two 16-row A tiles x four 16-col B tiles = 8 v8f accumulators).
// ---------------------------------------------------------------------------
template <int FIN>
__global__ __launch_bounds__(128) void gin_layer_kernel(
    const float* __restrict__ Xin, const float* __restrict__ Agg,
    const float* __restrict__ W1,  const float* __restrict__ bng,
    const float* __restrict__ bnb, const float* __restrict__ bnm,
    const float* __restrict__ bnv, const float* __restrict__ W2,
    float* __restrict__ Hout, int nNodes)
{
    constexpr int H  = HDIM;
    constexpr int HP = H + 4;     // padded k-stride for W2T / h1 rows
    constexpr int FP = FIN + 4;   // padded k-stride for W1T / x rows
    constexpr int NW = 4;         // waves per block

    extern __shared__ float smem[];
    float* sW1T   = smem;                    // H * FP   : W1 transposed, [n][k]
    float* sW2T   = sW1T + H * FP;           // H * HP   : W2 transposed, [n][k]
    float* sScale = sW2T + H * HP;           // H
    float* sOff   = sScale + H;              // H
    float* sXall  = sOff + H;                // NW * 32 * FP (reused for h1)

    const int t = threadIdx.x;

    // Stage W1/W2 transposed: B operand (W[kr][n], W[kr+1][n]) becomes one
    // aligned 8-byte LDS load per lane.
    for (int i = t; i < FIN * H; i += 128) {
        int k = i / H, n = i - k * H;
        sW1T[n * FP + k] = W1[i];
    }
    for (int i = t; i < H * H; i += 128) {
        int k = i / H, n = i - k * H;
        sW2T[n * HP + k] = W2[i];
    }
    if (t < H) {
        float sc  = bng[t] * rsqrtf(bnv[t] + 1e-5f);
        sScale[t] = sc;
        sOff[t]   = bnb[t] - bnm[t] * sc;
    }

    const int wave = t >> 5;      // wave32
    const int lane = t & 31;
    const int g    = lane >> 4;   // half-wave group (K-pair select)
    const int m    = lane & 15;   // row (A) / col (B,C,D) within 16x16 tile
    const int nodeBase = blockIdx.x * (NW * 32) + wave * 32;

    float* sx = sXall + wave * (32 * FP);

    // Stage (x + agg) 32 x FIN tile into LDS (float4-coalesced).
    for (int r = 0; r < 32; ++r) {
        int node = nodeBase + r;
        if (node > nNodes - 1) node = nNodes - 1;
        const float* xp = Xin + (size_t)node * FIN;
        const float* ap = Agg + (size_t)node * FIN;
        for (int c = lane * 4; c < FIN; c += 128) {
            float4 xv = *(const float4*)(xp + c);
            float4 av = *(const float4*)(ap + c);
            xv.x += av.x; xv.y += av.y; xv.z += av.z; xv.w += av.w;
            *(float4*)(sx + r * FP + c) = xv;
        }
    }
    __syncthreads();

    // ---- GEMM 1: h1 = (x+agg) @ W1, K = FIN ------------------------------
    v8f acc[8] = {};   // [rowTile(2)][colTile(4)] -> acc[rt*4 + ct]
    for (int k0 = 0; k0 < FIN; k0 += 4) {
        const int kr = k0 + 2 * g;
        const v2f aLo = *(const v2f*)(sx + m * FP + kr);         // rows 0..15
        const v2f aHi = *(const v2f*)(sx + (16 + m) * FP + kr);  // rows 16..31
        const v2f b0  = *(const v2f*)(sW1T + (m)      * FP + kr);
        const v2f b1  = *(const v2f*)(sW1T + (16 + m) * FP + kr);
        const v2f b2  = *(const v2f*)(sW1T + (32 + m) * FP + kr);
        const v2f b3  = *(const v2f*)(sW1T + (48 + m) * FP + kr);
        acc[0] = wmma4(aLo, b0, acc[0]);
        acc[1] = wmma4(aLo, b1, acc[1]);
        acc[2] = wmma4(aLo, b2, acc[2]);
        acc[3] = wmma4(aLo, b3, acc[3]);
        acc[4] = wmma4(aHi, b0, acc[4]);
        acc[5] = wmma4(aHi, b1, acc[5]);
        acc[6] = wmma4(aHi, b2, acc[6]);
        acc[7] = wmma4(aHi, b3, acc[7]);
    }

    // ---- BatchNorm (eval) + ReLU in accumulator layout -------------------
    const float sc0 = sScale[m],      of0 = sOff[m];
    const float sc1 = sScale[16 + m], of1 = sOff[16 + m];
    const float sc2 = sScale[32 + m], of2 = sOff[32 + m];
    const float sc3 = sScale[48 + m], of3 = sOff[48 + m];
#pragma unroll
    for (int rt = 0; rt < 2; ++rt) {
#pragma unroll
        for (int i = 0; i < 8; ++i) {
            acc[rt * 4 + 0][i] = fmaxf(acc[rt * 4 + 0][i] * sc0 + of0, 0.0f);
            acc[rt * 4 + 1][i] = fmaxf(acc[rt * 4 + 1][i] * sc1 + of1, 0.0f);
            acc[rt * 4 + 2][i] = fmaxf(acc[rt * 4 + 2][i] * sc2 + of2, 0.0f);
            acc[rt * 4 + 3][i] = fmaxf(acc[rt * 4 + 3][i] * sc3 + of3, 0.0f);
        }
    }

    // Restage h1 (32 x 64, row stride HP) into wave-private LDS so it can be
    // read back in the A-operand layout. (LDS ops are in-order per wave.)
#pragma unroll
    for (int rt = 0; rt < 2; ++rt) {
#pragma unroll
        for (int i = 0; i < 8; ++i) {
            const int r = rt * 16 + i + 8 * g;
            sx[r * HP + m]      = acc[rt * 4 + 0][i];
            sx[r * HP + 16 + m] = acc[rt * 4 + 1][i];
            sx[r * HP + 32 + m] = acc[rt * 4 + 2][i];
            sx[r * HP + 48 + m] = acc[rt * 4 + 3][i];
        }
    }

    // ---- GEMM 2: h2 = h1 @ W2, K = 64 ------------------------------------
    v8f d[8] = {};
    for (int k0 = 0; k0 < H; k0 += 4) {
        const int kr = k0 + 2 * g;
        const v2f aLo = *(const v2f*)(sx + m * HP + kr);
        const v2f aHi = *(const v2f*)(sx + (16 + m) * HP + kr);
        const v2f b0  = *(const v2f*)(sW2T + (m)      * HP + kr);
        const v2f b1  = *(const v2f*)(sW2T + (16 + m) * HP + kr);
        const v2f b2  = *(const v2f*)(sW2T + (32 + m) * HP + kr);
        const v2f b3  = *(const v2f*)(sW2T + (48 + m) * HP + kr);
        d[0] = wmma4(aLo, b0, d[0]);
        d[1] = wmma4(aLo, b1, d[1]);
        d[2] = wmma4(aLo, b2, d[2]);
        d[3] = wmma4(aLo, b3, d[3]);
        d[4] = wmma4(aHi, b0, d[4]);
        d[5] = wmma4(aHi, b1, d[5]);
        d[6] = wmma4(aHi, b2, d[6]);
        d[7] = wmma4(aHi, b3, d[7]);
    }

    // ---- ReLU + store (coalesced, guarded — all WMMA work done) ----------
#pragma unroll
    for (int rt = 0; rt < 2; ++rt) {
#pragma unroll
        for (int i = 0; i < 8; ++i) {
            const int row = nodeBase + rt * 16 + i + 8 * g;
            if (row < nNodes) {
                float* hp = Hout + (size_t)row * H;
                hp[m]      = fmaxf(d[rt * 4 + 0][i], 0.0f);
                hp[16 + m] = fmaxf(d[rt * 4 + 1][i], 0.0f);
                hp[32 + m] = fmaxf(d[rt * 4 + 2][i], 0.0f);
                hp[48 + m] = fmaxf(d[rt * 4 + 3][i], 0.0f);
            }
        }
    }
}

// ---------------------------------------------------------------------------
// Graph pooling: sums[batch[n]] += h[n], cnts[batch[n]] += 1
// ---------------------------------------------------------------------------
__global__ __launch_bounds__(256) void pool_kernel(
    const float* __restrict__ Hin, const int* __restrict__ batch,
    float* __restrict__ sums, float* __restrict__ cnts, int nNodes)
{
    long long tid = (long long)blockIdx.x * blockDim.x + threadIdx.x;
    if (tid >= (long long)nNodes * HDIM) return;
    int node = (int)(tid >> 6);
    int c    = (int)(tid & 63);
    int b    = batch[node];
    atomicAdd(sums + b * HDIM + c, Hin[tid]);
    if (c == 0) atomicAdd(cnts + b, 1.0f);
}

// ---------------------------------------------------------------------------
// Head: pooled mean -> relu(lin1) -> lin2 + bias -> log_softmax
// One block, one thread per graph (trivial FLOPs).
// ---------------------------------------------------------------------------
__global__ __launch_bounds__(128) void head_kernel(
    const float* __restrict__ sums, const float* __restrict__ cnts,
    const float* __restrict__ lin1, const float* __restrict__ lin2,
    const float* __restrict__ lin2b, float* __restrict__ out)
{
    __shared__ float sL1[HDIM * HDIM];
    __shared__ float sL2[HDIM * 10];
    __shared__ float sB[10];
    const int t = threadIdx.x;
    for (int i = t; i < HDIM * HDIM; i += 128) sL1[i] = lin1[i];
    for (int i = t; i < HDIM * 10;   i += 128) sL2[i] = lin2[i];
    if (t < 10) sB[t] = lin2b[t];
    __syncthreads();

    // t = graph index (128 graphs, 128 threads)
    const float cnt = fmaxf(cnts[t], 1.0f);
    float p[HDIM];
    for (int c = 0; c < HDIM; ++c) p[c] = sums[t * HDIM + c] / cnt;

    float z[10];
    for (int j = 0; j < 10; ++j) z[j] = sB[j];
    for (int k = 0; k < HDIM; ++k) {
        float a = 0.0f;
        for (int c = 0; c < HDIM; ++c) a += p[c] * sL1[c * HDIM + k];
        a = fmaxf(a, 0.0f);                 // relu(pooled @ lin1)
        for (int j = 0; j < 10; ++j) z[j] += a * sL2[k * 10 + j];
    }
    float mx = z[0];
    for (int j = 1; j < 10; ++j) mx = fmaxf(mx, z[j]);
    float s = 0.0f;
    for (int j = 0; j < 10; ++j) s += expf(z[j] - mx);
    const float lse = logf(s);
    for (int j = 0; j < 10; ++j) out[t * 10 + j] = z[j] - mx - lse;
}

// ---------------------------------------------------------------------------
extern "C" void kernel_launch(void* const* d_in, const int* in_sizes, int n_in,
                              void* d_out, int out_size, void* d_ws, size_t ws_size,
                              hipStream_t stream)
{
    (void)in_sizes; (void)n_in; (void)out_size; (void)ws_size;

    const float* x          = (const float*)d_in[0];
    const int*   edge_index = (const int*)d_in[1];   // [2, E]
    const int*   batch      = (const int*)d_in[2];
    const float* W1_0  = (const float*)d_in[3];
    const float* bng0  = (const float*)d_in[4];
    const float* bnb0  = (const float*)d_in[5];
    const float* bnm0  = (const float*)d_in[6];
    const float* bnv0  = (const float*)d_in[7];
    const float* W2_0  = (const float*)d_in[8];
    const float* W1_r  = (const float*)d_in[9];      // [3,64,64]
    const float* bngr  = (const float*)d_in[10];
    const float* bnbr  = (const float*)d_in[11];
    const float* bnmr  = (const float*)d_in[12];
    const float* bnvr  = (const float*)d_in[13];
    const float* W2_r  = (const float*)d_in[14];
    const float* lin1  = (const float*)d_in[15];
    const float* lin2  = (const float*)d_in[16];
    const float* lin2b = (const float*)d_in[17];

    const int* src = edge_index;               // row 0
    const int* dst = edge_index + N_EDGES_C;   // row 1

    // Workspace partition (~103 MB)
    float* agg  = (float*)d_ws;
    float* hA   = agg + (size_t)N_NODES_C * 128;
    float* hB   = hA  + (size_t)N_NODES_C * HDIM;
    float* pool = hB  + (size_t)N_NODES_C * HDIM;   // 128*64 sums + 128 cnts
    float* cnt  = pool + N_GRAPHS_C * HDIM;

    constexpr int HP = HDIM + 4;
    const int ginGrid = (N_NODES_C + 127) / 128;    // 128 nodes per block

    // ---- Layer 0 (F=128 -> 64) ----
    hipMemsetAsync(agg, 0, (size_t)N_NODES_C * 128 * sizeof(float), stream);
    {
        long long total = (long long)N_EDGES_C * (128 / 4);
        int blocks = (int)((total + 255) / 256);
        scatter_add_kernel<128><<<blocks, 256, 0, stream>>>(x, src, dst, agg);
    }
    {
        constexpr int FP0 = 128 + 4;
        size_t shmem = (size_t)(HDIM * FP0 /*sW1T*/ + HDIM * HP /*sW2T*/
                                + 2 * HDIM + 4 * 32 * FP0) * sizeof(float);
        gin_layer_kernel<128><<<ginGrid, 128, shmem, stream>>>(
            x, agg, W1_0, bng0, bnb0, bnm0, bnv0, W2_0, hA, N_NODES_C);
    }

    // ---- Layers 1..3 (64 -> 64) ----
    const float* hin = hA;
    float* hout = hB;
    for (int i = 0; i < 3; ++i) {
        hipMemsetAsync(agg, 0, (size_t)N_NODES_C * HDIM * sizeof(float), stream);
        long long total = (long long)N_EDGES_C * (HDIM / 4);
        int blocks = (int)((total + 255) / 256);
        scatter_add_kernel<HDIM><<<blocks, 256, 0, stream>>>(hin, src, dst, agg);

        size_t shmem = (size_t)(HDIM * HP /*sW1T*/ + HDIM * HP /*sW2T*/
                                + 2 * HDIM + 4 * 32 * HP) * sizeof(float);
        gin_layer_kernel<HDIM><<<ginGrid, 128, shmem, stream>>>(
            hin, agg,
            W1_r + (size_t)i * HDIM * HDIM,
            bngr + (size_t)i * HDIM, bnbr + (size_t)i * HDIM,
            bnmr + (size_t)i * HDIM, bnvr + (size_t)i * HDIM,
            W2_r + (size_t)i * HDIM * HDIM,
            hout, N_NODES_C);

        const float* tmp = hin; hin = hout; hout = (float*)tmp;
    }

    // ---- Pool + head ----
    hipMemsetAsync(pool, 0, (size_t)(N_GRAPHS_C * HDIM + N_GRAPHS_C) * sizeof(float), stream);
    {
        long long total = (long long)N_NODES_C * HDIM;
        int blocks = (int)((total + 255) / 256);
        pool_kernel<<<blocks, 256, 0, stream>>>(hin, batch, pool, cnt, N_NODES_C);
    }
    head_kernel<<<1, 128, 0, stream>>>(pool, cnt, lin1, lin2, lin2b, (float*)d_out);
}